// Group_532575945286
// MI455X (gfx1250) — compile-verified
//
#include <hip/hip_runtime.h>
#include <hip/hip_bf16.h>

// Problem constants (match reference)
#define BATCH      16
#define N_POINTS   8192
#define FEAT_C     384
#define NUM_GROUP  512
#define GROUP_M    32

// FPS kernel config
#define FPS_THREADS 1024
#define NPT (N_POINTS / FPS_THREADS)   // 8 points per thread

// KNN kernel config
#define QT        16     // queries per block (one WMMA M tile)
#define PCHUNK    512    // points per LDS distance-tile chunk
#define K2_THREADS 512   // 16 waves of 32

typedef __attribute__((ext_vector_type(2))) float v2f;
typedef __attribute__((ext_vector_type(8))) float v8f;

// ---- Tensor Data Mover availability / arity detection -----------------------
#if __has_include(<hip/amd_detail/amd_gfx1250_TDM.h>)
#define TDM_6ARG 1
#endif
#if __has_builtin(__builtin_amdgcn_tensor_load_to_lds)
#define HAVE_TDM 1
#endif

typedef unsigned int u32x4 __attribute__((ext_vector_type(4)));
typedef int          i32x4 __attribute__((ext_vector_type(4)));
typedef int          i32x8 __attribute__((ext_vector_type(8)));

#ifdef HAVE_TDM
// Issue one TDM load of `nelem` contiguous f32 from global -> LDS byte offset.
// 1-D tensor: D# group0 = {count=1, lds_addr, global_addr, type=2},
// group1 = {data_size=4B, tensor_dim0 = tile_dim0 = nelem}; groups 2/3 zero.
__device__ __forceinline__ void tdm_load_f32(const float* gsrc,
                                             unsigned lds_byte_off, int nelem) {
    unsigned long long ga = (unsigned long long)(uintptr_t)gsrc;
    u32x4 g0;
    g0.x = 1u;                                              // count=1, is_restore=0
    g0.y = lds_byte_off;                                    // lds_addr [63:32]
    g0.z = (unsigned)(ga & 0xFFFFFFFFu);                    // global_addr low
    g0.w = (unsigned)((ga >> 32) & 0x01FFFFFFu) | 0x80000000u; // addr[56:32] | type=2

    i32x8 g1;
    g1[0] = 0x00020000;                                     // data_size = 4B
    g1[1] = (nelem & 0xFFFF) << 16;                         // tensor_dim0[15:0]
    g1[2] = (nelem >> 16) & 0xFFFF;                         // tensor_dim0[31:16]
    g1[3] = (nelem & 0xFFFF) << 16;                         // tile_dim0
    g1[4] = 0;                                              // tile_dim1/2 unused
    g1[5] = nelem;                                          // tensor_dim0_stride low
    g1[6] = 0;
    g1[7] = 0;

    i32x4 z4 = {0, 0, 0, 0};
#ifdef TDM_6ARG
    i32x8 z8 = {0, 0, 0, 0, 0, 0, 0, 0};
    __builtin_amdgcn_tensor_load_to_lds(g0, g1, z4, z4, z8, 0);
#else
    __builtin_amdgcn_tensor_load_to_lds(g0, g1, z4, z4, 0);
#endif
}
#endif

// ---------------------------------------------------------------------------
// Wave32 bitonic compare-exchange on (key,value) pairs across lanes.
// asc_block: within this bitonic block, lower lane keeps min (ascending).
// ---------------------------------------------------------------------------
__device__ __forceinline__ void cmp_xchg(float& d, int& i, int j, bool asc_block) {
    float od = __shfl_xor(d, j, 32);
    int   oi = __shfl_xor(i, j, 32);
    bool lower    = ((threadIdx.x & (unsigned)j) == 0u);
    bool take_min = (lower == asc_block);
    bool repl     = take_min ? (od < d) : (od > d);
    if (repl) { d = od; i = oi; }
}

// ===========================================================================
// Kernel 1: Farthest Point Sampling. One block per batch; 512 dependent
// steps; per step a full-block max-reduction over the running min-distances.
// Writes the 512 center coordinates straight into the d_out center section.
// ===========================================================================
__global__ void __launch_bounds__(FPS_THREADS)
fps_kernel(const float* __restrict__ xyz, float* __restrict__ center_out) {
    const int b   = blockIdx.x;
    const int tid = threadIdx.x;
    const float* x = xyz + (size_t)b * N_POINTS * 3;

    float px[NPT], py[NPT], pz[NPT], md[NPT];
#pragma unroll
    for (int k = 0; k < NPT; ++k) {
        int p = tid + k * FPS_THREADS;
        px[k] = x[p * 3 + 0];
        py[k] = x[p * 3 + 1];
        pz[k] = x[p * 3 + 2];
        md[k] = 1.0e10f;
    }

    __shared__ float s_cen[3];
    __shared__ int   s_choice;
    __shared__ float s_wd[32];
    __shared__ int   s_wi[32];
    if (tid == 0) s_choice = 0;   // deterministic start at index 0
    __syncthreads();

    for (int g = 0; g < NUM_GROUP; ++g) {
        if (tid == 0) {
            int f = s_choice;
            float cx = x[f * 3 + 0], cy = x[f * 3 + 1], cz = x[f * 3 + 2];
            s_cen[0] = cx; s_cen[1] = cy; s_cen[2] = cz;
            float* co = center_out + ((size_t)b * NUM_GROUP + g) * 3;
            co[0] = cx; co[1] = cy; co[2] = cz;
        }
        __syncthreads();   // A: s_cen valid

        float cx = s_cen[0], cy = s_cen[1], cz = s_cen[2];
        float bd = -1.0f; int bi = 0;
#pragma unroll
        for (int k = 0; k < NPT; ++k) {
            float dx = px[k] - cx, dy = py[k] - cy, dz = pz[k] - cz;
            float d  = dx * dx + dy * dy + dz * dz;
            md[k] = fminf(md[k], d);
            if (md[k] > bd) { bd = md[k]; bi = tid + k * FPS_THREADS; }
        }
        // wave32 max-reduce (value,index)
#pragma unroll
        for (int off = 16; off >= 1; off >>= 1) {
            float od = __shfl_xor(bd, off, 32);
            int   oi = __shfl_xor(bi, off, 32);
            if (od > bd) { bd = od; bi = oi; }
        }
        int wv = tid >> 5;
        if ((tid & 31) == 0) { s_wd[wv] = bd; s_wi[wv] = bi; }
        __syncthreads();   // B: s_wd valid

        if (tid == 0) {
            float best = s_wd[0]; int bidx = s_wi[0];
#pragma unroll
            for (int w = 1; w < 32; ++w)
                if (s_wd[w] > best) { best = s_wd[w]; bidx = s_wi[w]; }
            s_choice = bidx;
        }
        __syncthreads();   // C: s_choice valid
    }
}

// ===========================================================================
// Kernel 2: TDM-staged KNN via WMMA distance tiles + wave bitonic top-32,
// then gather. Block = 16 queries x 16 waves. Per 512-point chunk:
//   (1) wave 0 TDM-loads chunk xyz (6 KB) into LDS, s_wait_tensorcnt
//   (2) distance tile d = |q|^2 + |p|^2 - 2 q.p as one V_WMMA_F32_16X16X4_F32
//       A[m] = (qx,qy,qz,|q|^2), B[:,n] = (-2px,-2py,-2pz,1), C[m,n] = |p_n|^2
//   (3) wave w streams row w, maintains top-32 via bitonic merge network
// ===========================================================================
__global__ void __launch_bounds__(K2_THREADS)
knn_group_kernel(const float* __restrict__ xyz,
                 const float* __restrict__ features,
                 const float* __restrict__ center,
                 float* __restrict__ out_nb,
                 float* __restrict__ out_fg) {
    const int blocks_per_batch = NUM_GROUP / QT;
    const int b    = blockIdx.x / blocks_per_batch;
    const int qt   = blockIdx.x % blocks_per_batch;
    const int tid  = threadIdx.x;
    const int lane = tid & 31;
    const int wave = tid >> 5;
    const int m    = lane & 15;

    __shared__ float sdist[QT * PCHUNK];   // 32 KB distance tile
    __shared__ float sxyz[PCHUNK * 3];     // 6 KB staged chunk xyz
    __shared__ float sq[QT * 4];           // qx,qy,qz,|q|^2 per query

    const float* xb = xyz + (size_t)b * N_POINTS * 3;

    if (tid < QT) {
        const float* c = center + ((size_t)b * NUM_GROUP + qt * QT + tid) * 3;
        float qx = c[0], qy = c[1], qz = c[2];
        sq[tid * 4 + 0] = qx;
        sq[tid * 4 + 1] = qy;
        sq[tid * 4 + 2] = qz;
        sq[tid * 4 + 3] = qx * qx + qy * qy + qz * qz;
    }
    __syncthreads();

    // A fragment: lanes 0-15 hold (K0,K1)=(qx,qy); lanes 16-31 hold (K2,K3)=(qz,|q|^2)
    v2f afrag;
    afrag.x = (lane < 16) ? sq[m * 4 + 0] : sq[m * 4 + 2];
    afrag.y = (lane < 16) ? sq[m * 4 + 1] : sq[m * 4 + 3];

    // Running top-32 for this wave's query, sorted ascending across lanes.
    float val = 3.0e38f;
    int   idx = 0;

    for (int base = 0; base < N_POINTS; base += PCHUNK) {
        // ---- Stage chunk xyz into LDS -------------------------------------
#ifdef HAVE_TDM
        if (wave == 0) {
            tdm_load_f32(xb + (size_t)base * 3,
                         (unsigned)(uintptr_t)(void*)sxyz, PCHUNK * 3);
            __builtin_amdgcn_s_wait_tensorcnt(0);   // DMA complete before barrier
        }
#else
        for (int i = tid; i < PCHUNK * 3; i += K2_THREADS)
            sxyz[i] = xb[(size_t)base * 3 + i];
#endif
        __syncthreads();

        // ---- Phase A: WMMA distance tiles into LDS (wave w -> cols w*32..+31)
#pragma unroll
        for (int t = 0; t < 2; ++t) {
            int col = wave * 32 + t * 16 + m;   // column within chunk
            float px = sxyz[col * 3 + 0];
            float py = sxyz[col * 3 + 1];
            float pz = sxyz[col * 3 + 2];
            v2f bfrag;
            bfrag.x = (lane < 16) ? (-2.0f * px) : (-2.0f * pz);
            bfrag.y = (lane < 16) ? (-2.0f * py) : 1.0f;
            float s = px * px + py * py + pz * pz;
            v8f cfrag;
#pragma unroll
            for (int r = 0; r < 8; ++r) cfrag[r] = s;

            v8f d = __builtin_amdgcn_wmma_f32_16x16x4_f32(
                /*neg_a=*/false, afrag, /*neg_b=*/false, bfrag,
                /*c_mod=*/(short)0, cfrag, /*reuse_a=*/false, /*reuse_b=*/false);

            int mrow0 = (lane < 16) ? 0 : 8;   // D layout: VGPR r -> row r (+8 upper half)
#pragma unroll
            for (int r = 0; r < 8; ++r)
                sdist[(mrow0 + r) * PCHUNK + col] = d[r];
        }
        __syncthreads();

        // ---- Phase B: wave w streams row w, maintains top-32 --------------
#pragma unroll 1
        for (int cs = 0; cs < PCHUNK / 32; ++cs) {
            int   col  = cs * 32 + lane;
            float dnew = sdist[wave * PCHUNK + col];
            int   pnew = base + col;

            float thr = __shfl(val, 31, 32);   // current 32nd-smallest (wave-uniform)
            if (__any(dnew < thr)) {
                // bitonic sort the 32 new candidates DESCENDING across lanes
#pragma unroll
                for (int k = 2; k <= 32; k <<= 1) {
#pragma unroll
                    for (int j = k >> 1; j >= 1; j >>= 1) {
                        bool asc = ((lane & k) != 0);
                        cmp_xchg(dnew, pnew, j, asc);
                    }
                }
                // elementwise min against ascending list -> 32 smallest (bitonic)
                if (dnew < val) { val = dnew; idx = pnew; }
                // bitonic clean back to ascending
#pragma unroll
                for (int j = 16; j >= 1; j >>= 1)
                    cmp_xchg(val, idx, j, true);
            }
        }
        __syncthreads();   // sdist/sxyz reused next chunk
    }

    // ---- Output: wave w owns query g = qt*QT + w; lane k holds k-th nearest
    const int g = qt * QT + wave;
    const size_t qbase = (size_t)b * NUM_GROUP + g;
    const float cx = sq[wave * 4 + 0];
    const float cy = sq[wave * 4 + 1];
    const float cz = sq[wave * 4 + 2];

    {   // neighborhood = xyz[idx] - center
        int p = idx;
        float nx = xb[p * 3 + 0] - cx;
        float ny = xb[p * 3 + 1] - cy;
        float nz = xb[p * 3 + 2] - cz;
        float* o = out_nb + (qbase * GROUP_M + lane) * 3;
        o[0] = nx; o[1] = ny; o[2] = nz;
    }

    // feature gather: 32 rows of 384 floats, float4 lanes
    const float* fb = features + (size_t)b * N_POINTS * FEAT_C;
#pragma unroll 1
    for (int k = 0; k < GROUP_M; ++k) {
        int p = __shfl(idx, k, 32);
        if (k + 1 < GROUP_M) {
            int pn = __shfl(idx, k + 1, 32);
            __builtin_prefetch(fb + (size_t)pn * FEAT_C, 0, 0);  // global_prefetch_b8
        }
        const float4* src = (const float4*)(fb + (size_t)p * FEAT_C);
        float4*       dst = (float4*)(out_fg + (qbase * GROUP_M + k) * FEAT_C);
#pragma unroll
        for (int c = lane; c < FEAT_C / 4; c += 32)
            dst[c] = src[c];
    }
}

// ===========================================================================
extern "C" void kernel_launch(void* const* d_in, const int* in_sizes, int n_in,
                              void* d_out, int out_size, void* d_ws, size_t ws_size,
                              hipStream_t stream) {
    (void)in_sizes; (void)n_in; (void)out_size; (void)d_ws; (void)ws_size;
    const float* xyz      = (const float*)d_in[0];
    const float* features = (const float*)d_in[1];

    float* out        = (float*)d_out;
    float* out_nb     = out;                                              // [B,G,M,3]
    float* out_center = out_nb + (size_t)BATCH * NUM_GROUP * GROUP_M * 3; // [B,G,3]
    float* out_fg     = out_center + (size_t)BATCH * NUM_GROUP * 3;       // [B,G,M,C]

    fps_kernel<<<BATCH, FPS_THREADS, 0, stream>>>(xyz, out_center);

    dim3 grid(BATCH * (NUM_GROUP / QT));
    knn_group_kernel<<<grid, K2_THREADS, 0, stream>>>(
        xyz, features, out_center, out_nb, out_fg);
}